// MatrixVectorScaledDotProductAttention_33466385171053
// MI455X (gfx1250) — compile-verified
//
#include <hip/hip_runtime.h>
#include <math.h>

typedef __attribute__((ext_vector_type(2))) float v2f;
typedef __attribute__((ext_vector_type(4))) float v4f;
typedef __attribute__((ext_vector_type(8))) float v8f;

#define DHEAD   128
#define KSTRIDE 132              // LDS row stride in dwords (conflict-free, 16B rows)
#define TILEF   (16 * KSTRIDE)   // floats per K tile in LDS
#define ROWF    132              // ws row stride in floats: [m, l, pad, pad, acc0..127]
#define TEMP    11.313708498984761f

// ---------------------------------------------------------------------------
// Pass 1: per (batch, segment) online-softmax partial over a key range.
// One wave per workgroup. K tiles are streamed global->LDS with
// GLOBAL_LOAD_ASYNC_TO_LDS_B128 (double-buffered, ASYNCcnt), scores via
// V_WMMA_F32_16X16X4_F32, V streamed with non-temporal b128 loads.
// ---------------------------------------------------------------------------
__global__ __launch_bounds__(32) void attn_partial_kernel(
    const float* __restrict__ q, const float* __restrict__ k,
    const float* __restrict__ v, const unsigned char* __restrict__ mask,
    float* __restrict__ attn_raw, float* __restrict__ stats,
    int N, int S, int nkps)
{
  const int wg   = blockIdx.x;
  const int b    = wg / S;
  const int seg  = wg % S;
  const int lane = threadIdx.x;
  const int n0   = seg * nkps;
  const int n1   = min(N, n0 + nkps);
  const int nt   = (n1 - n0) >> 4;          // full 16-key tiles (N % 16 == 0)

  __shared__ float qs[DHEAD];
  __shared__ float kt[2][TILEF];
  __shared__ float st[16];

  // q row -> LDS (one b128 per lane)
  *(v4f*)&qs[lane * 4] = *(const v4f*)(q + (size_t)b * DHEAD + lane * 4);
  __syncthreads();

  const float invT = 1.0f / TEMP;
  const float* kb = k + (size_t)b * N * DHEAD;
  const float* vb = v + (size_t)b * N * DHEAD;
  const unsigned char* mb = mask + (size_t)b * N;
  float* ab = attn_raw + (size_t)b * N;

  // LDS byte offset of kt: low 32 bits of the generic pointer are the LDS addr.
  const unsigned kt_lds = (unsigned)(size_t)&kt[0][0];

  float m = -INFINITY;
  float l = 0.0f;
  v4f acc = {0.f, 0.f, 0.f, 0.f};           // lane owns output dims [4*lane, 4*lane+4)

  const int half = lane >> 4;               // 0: K-dims {4c,4c+1}; 1: {4c+2,4c+3}
  const int km   = lane & 15;               // A-operand key row

  // Issue one 16x128 K tile as 16 per-lane async b128 copies into LDS buffer.
  auto issue_tile = [&](int buf, int n) {
    const float* base = kb + (size_t)n * DHEAD;            // wave-uniform SGPR pair
    #pragma unroll
    for (int i = 0; i < 16; ++i) {
      unsigned loff = kt_lds + (unsigned)(buf * (TILEF * 4) + i * (KSTRIDE * 4) + lane * 16);
      unsigned goff = (unsigned)(i * DHEAD * 4 + lane * 16);
      asm volatile("global_load_async_to_lds_b128 %0, %1, %2"
                   :: "v"(loff), "v"(goff), "s"(base) : "memory");
    }
  };

  if (nt > 0) issue_tile(0, n0);

  for (int t = 0; t < nt; ++t) {
    const int n = n0 + t * 16;

    // Prefetch next tile into the other buffer, then wait for current tile.
    if (t + 1 < nt) {
      issue_tile((t + 1) & 1, n + 16);
      asm volatile("s_wait_asynccnt 16" ::: "memory");   // current tile landed
    } else {
      asm volatile("s_wait_asynccnt 0" ::: "memory");
    }
    __syncthreads();

    const float* ktile = kt[t & 1];

    // Scores for 16 keys: D += A(16x4 of K-tile) x B(4x16 = q chunk bcast).
    v8f d = {0.f, 0.f, 0.f, 0.f, 0.f, 0.f, 0.f, 0.f};
    #pragma unroll
    for (int c = 0; c < 32; ++c) {
      v2f a  = *(const v2f*)&ktile[km * KSTRIDE + 4 * c + half * 2];
      v2f bq = *(const v2f*)&qs[4 * c + half * 2];
      d = __builtin_amdgcn_wmma_f32_16x16x4_f32(false, a, false, bq,
                                                (short)0, d, false, false);
    }

    // Column 0 of D: lane 0 holds keys 0-7, lane 16 holds keys 8-15.
    if (km == 0) {
      #pragma unroll
      for (int r = 0; r < 8; ++r) st[half * 8 + r] = d[r];
    }
    __syncthreads();

    // Online-softmax update (all lanes replicate the 16 scalars).
    float s[16];
    float tmax = -INFINITY;
    #pragma unroll
    for (int j = 0; j < 16; ++j) {
      float sc = st[j] * invT;
      if (__builtin_nontemporal_load(mb + n + j)) sc = -INFINITY;
      s[j] = sc;
      tmax = fmaxf(tmax, sc);
    }
    if (lane < 16) ab[n + lane] = s[lane];   // raw masked/scaled score

    float mnew  = fmaxf(m, tmax);
    float scale = __expf(m - mnew);
    l *= scale;
    acc *= scale;
    #pragma unroll
    for (int j = 0; j < 16; ++j) {
      float p = __expf(s[j] - mnew);
      l += p;
      v4f vv = __builtin_nontemporal_load(
                   (const v4f*)(vb + (size_t)(n + j) * DHEAD + lane * 4));
      acc += p * vv;                          // coalesced, streamed-once V
    }
    m = mnew;
    __syncthreads();
  }

  float* row = stats + (size_t)(b * S + seg) * ROWF;
  if (lane == 0) { row[0] = m; row[1] = l; }
  *(v4f*)&row[4 + lane * 4] = acc;
}

// ---------------------------------------------------------------------------
// Pass 2: per batch row, merge segment stats -> output, normalize attn in place.
// ---------------------------------------------------------------------------
__global__ __launch_bounds__(128) void attn_finalize_kernel(
    const float* __restrict__ stats, float* __restrict__ out,
    float* __restrict__ attn, int N, int S)
{
  const int b = blockIdx.x;
  const int t = threadIdx.x;
  const float* srow = stats + (size_t)b * S * ROWF;

  float M = -INFINITY;
  for (int s = 0; s < S; ++s) M = fmaxf(M, srow[s * ROWF + 0]);
  float L = 0.f;
  for (int s = 0; s < S; ++s) {
    float ms = srow[s * ROWF + 0];
    float ls = srow[s * ROWF + 1];
    if (ls > 0.f) L += ls * __expf(ms - M);
  }

  float o = 0.f;
  for (int s = 0; s < S; ++s) {
    float w = __expf(srow[s * ROWF + 0] - M);
    if (srow[s * ROWF + 1] > 0.f)
      o += srow[s * ROWF + 4 + t] * w;
  }
  float invL = 1.0f / L;
  out[(size_t)b * DHEAD + t] = o * invL;

  float* arow = attn + (size_t)b * N;
  for (int n = t; n < N; n += 128)
    arow[n] = __expf(arow[n] - M) * invL;
}

// ---------------------------------------------------------------------------
extern "C" void kernel_launch(void* const* d_in, const int* in_sizes, int n_in,
                              void* d_out, int out_size, void* d_ws, size_t ws_size,
                              hipStream_t stream)
{
  const float*         q    = (const float*)d_in[0];
  const float*         k    = (const float*)d_in[1];
  const float*         v    = (const float*)d_in[2];
  const unsigned char* mask = (const unsigned char*)d_in[3];

  const int B = in_sizes[0] / DHEAD;            // 64
  const int N = in_sizes[1] / (B * DHEAD);      // 8192

  float* out   = (float*)d_out;                         // [B, 128]
  float* attn  = (float*)d_out + (size_t)B * DHEAD;     // [B, N]
  float* stats = (float*)d_ws;

  // Segments per batch, limited by workspace (B*S rows of ROWF floats).
  int S = (int)(ws_size / ((size_t)B * ROWF * sizeof(float)));
  if (S < 1)  S = 1;
  if (S > 32) S = 32;
  int nkps = ((N + S - 1) / S + 15) & ~15;      // keys per segment, mult of 16

  attn_partial_kernel<<<B * S, 32, 0, stream>>>(q, k, v, mask, attn, stats,
                                                N, S, nkps);
  attn_finalize_kernel<<<B, 128, 0, stream>>>(stats, out, attn, N, S);
}